// PureSurface_59322088292652
// MI455X (gfx1250) — compile-verified
//
#include <hip/hip_runtime.h>
#include <math.h>

// Problem geometry: (8, 1, 512, 512) float32 / int32
#define BIG   1e8f
#define NB    8
#define NLINE 512
#define HW    262144      // 512*512
#define NTOT  2097152     // 8*512*512

typedef float v2f __attribute__((ext_vector_type(2)));
typedef float v8f __attribute__((ext_vector_type(8)));

// ---------------------------------------------------------------------------
// Kernel 1: f = (y != 0 ? 0 : BIG), then exact 1-D squared-distance transform
// along the batch axis (n = 8), fully register-resident per (h,w) pixel.
// ---------------------------------------------------------------------------
__global__ __launch_bounds__(256) void k_init_dt_b(const int* __restrict__ y,
                                                   float* __restrict__ f) {
  int t = blockIdx.x * 256 + threadIdx.x;   // 0 .. HW-1, coalesced per batch plane
  float v[NB];
#pragma unroll
  for (int b = 0; b < NB; ++b)
    v[b] = (y[(size_t)b * HW + t] != 0) ? 0.0f : BIG;
#pragma unroll
  for (int i = 0; i < NB; ++i) {
    float d = BIG;
#pragma unroll
    for (int j = 0; j < NB; ++j) {
      float dj = (float)(i - j);
      d = fminf(d, fmaf(dj, dj, v[j]));
    }
    f[(size_t)i * HW + t] = d;
  }
}

// ---------------------------------------------------------------------------
// Kernel 2: exact DT along w (contiguous lines of 512). One block per line,
// line staged in LDS, each thread produces 2 outputs. Inner loop reads LDS as
// float4 (ds_load_b128): one DS op feeds 8 fma + 8 min (dual-issue friendly).
// In-place safe: all reads happen from LDS after the barrier.
// ---------------------------------------------------------------------------
__global__ __launch_bounds__(256) void k_dt_w(float* __restrict__ f) {
  __shared__ float s[NLINE];
  float* base = f + (size_t)blockIdx.x * NLINE;   // blockIdx.x in [0, 8*512)
  int t = threadIdx.x;
  s[t]       = base[t];
  s[t + 256] = base[t + 256];
  __syncthreads();
  const float4* s4 = (const float4*)s;
  float d0 = BIG, d1 = BIG;
  float x0 = (float)t;
  float x1 = (float)(t + 256);
#pragma unroll 2
  for (int q = 0; q < NLINE / 4; ++q) {
    float4 sj = s4[q];                     // j = 4q .. 4q+3
    d0 = fminf(d0, fmaf(x0, x0, sj.x));
    d1 = fminf(d1, fmaf(x1, x1, sj.x));
    x0 -= 1.0f; x1 -= 1.0f;
    d0 = fminf(d0, fmaf(x0, x0, sj.y));
    d1 = fminf(d1, fmaf(x1, x1, sj.y));
    x0 -= 1.0f; x1 -= 1.0f;
    d0 = fminf(d0, fmaf(x0, x0, sj.z));
    d1 = fminf(d1, fmaf(x1, x1, sj.z));
    x0 -= 1.0f; x1 -= 1.0f;
    d0 = fminf(d0, fmaf(x0, x0, sj.w));
    d1 = fminf(d1, fmaf(x1, x1, sj.w));
    x0 -= 1.0f; x1 -= 1.0f;
  }
  base[t]       = d0;
  base[t + 256] = d1;
}

// ---------------------------------------------------------------------------
// Kernel 3: exact DT along h (stride-512 lines). Each block owns a full-height
// 512 x 16 column tile of one batch plane in 32 KB LDS (coalesced 64B global
// segments, conflict-free LDS: lanes 0-15 consecutive words, 16-31 broadcast).
// 4 accumulators per j-iteration amortize the LDS read (1 DS : 12 VALU).
// ---------------------------------------------------------------------------
#define TW 16
__global__ __launch_bounds__(256) void k_dt_h(float* __restrict__ f) {
  __shared__ float s[NLINE * TW];          // 32 KB
  int b  = blockIdx.y;
  int w0 = blockIdx.x * TW;
  int t  = threadIdx.x;
  int w  = t & (TW - 1);
  int r  = t >> 4;                         // 0..15
  float* base = f + (size_t)b * HW + w0;
#pragma unroll
  for (int p = 0; p < 32; ++p) {
    int h = r + 16 * p;
    s[h * TW + w] = base[(size_t)h * NLINE + w];
  }
  __syncthreads();
  for (int pp = 0; pp < 8; ++pp) {
    int ib = r + 64 * pp;
    float d0 = BIG, d1 = BIG, d2 = BIG, d3 = BIG;
    float x0 = (float)ib, x1 = x0 + 16.f, x2 = x0 + 32.f, x3 = x0 + 48.f;
#pragma unroll 4
    for (int j = 0; j < NLINE; ++j) {
      float sj = s[j * TW + w];
      d0 = fminf(d0, fmaf(x0, x0, sj));
      d1 = fminf(d1, fmaf(x1, x1, sj));
      d2 = fminf(d2, fmaf(x2, x2, sj));
      d3 = fminf(d3, fmaf(x3, x3, sj));
      x0 -= 1.f; x1 -= 1.f; x2 -= 1.f; x3 -= 1.f;
    }
    base[(size_t)ib        * NLINE + w] = d0;
    base[(size_t)(ib + 16) * NLINE + w] = d1;
    base[(size_t)(ib + 32) * NLINE + w] = d2;
    base[(size_t)(ib + 48) * NLINE + w] = d3;
  }
}

// ---------------------------------------------------------------------------
// Kernel 4: partial = sum(sigmoid(x) * sqrt(dist)) via V_WMMA_F32_16X16X4_F32.
// With B = ones, D[i][n] = sum_k A[i][k] + C: every loaded element contributes
// 16x to the D tile, so block total = LDS-reduce(per-lane D sums) / 16.
// Sigmoid via v_exp_f32 + v_rcp_f32 (no IEEE div sequence).
// Uniform control flow before/at every WMMA => EXEC all-ones as required.
// Grid: 512 blocks x 256 threads = 131072 lanes; 8 iters x 2 elem/lane = 2^21.
// ---------------------------------------------------------------------------
__global__ __launch_bounds__(256) void k_reduce_wmma(const float* __restrict__ x,
                                                     const float* __restrict__ d,
                                                     float* __restrict__ partials) {
  const int S = 512 * 256;                 // total lanes in grid
  int gid = blockIdx.x * 256 + threadIdx.x;
  v8f c = {};
  v2f bones; bones[0] = 1.0f; bones[1] = 1.0f;
  for (int it = 0; it < 8; ++it) {
    int i0 = gid + it * S;                 // [0, 2^20)  coalesced
    int i1 = i0 + (NTOT / 2);              // [2^20, 2^21) coalesced
    // sigmoid(x)*sqrt(dist) = sqrt(d) * rcp(1 + exp(-x))
    float a0 = sqrtf(d[i0]) * __builtin_amdgcn_rcpf(1.0f + __expf(-x[i0]));
    float a1 = sqrtf(d[i1]) * __builtin_amdgcn_rcpf(1.0f + __expf(-x[i1]));
    v2f a; a[0] = a0; a[1] = a1;
    // 8 args: (neg_a, A, neg_b, B, c_mod, C, reuse_a, reuse_b)
    c = __builtin_amdgcn_wmma_f32_16x16x4_f32(false, a, false, bones,
                                              (short)0, c, false, false);
  }
  __shared__ float red[256];
  red[threadIdx.x] = c[0] + c[1] + c[2] + c[3] + c[4] + c[5] + c[6] + c[7];
  __syncthreads();
#pragma unroll
  for (int o = 128; o > 0; o >>= 1) {
    if (threadIdx.x < o) red[threadIdx.x] += red[threadIdx.x + o];
    __syncthreads();
  }
  if (threadIdx.x == 0) partials[blockIdx.x] = red[0] * (1.0f / 16.0f);
}

// ---------------------------------------------------------------------------
// Kernel 5: fold 512 block partials into the final mean.
// ---------------------------------------------------------------------------
__global__ __launch_bounds__(256) void k_final(const float* __restrict__ partials,
                                               float* __restrict__ out) {
  __shared__ float s[256];
  int t = threadIdx.x;
  s[t] = partials[t] + partials[t + 256];
  __syncthreads();
  for (int o = 128; o > 0; o >>= 1) {
    if (t < o) s[t] += s[t + o];
    __syncthreads();
  }
  if (t == 0) out[0] = s[0] * (1.0f / (float)NTOT);
}

// ---------------------------------------------------------------------------
extern "C" void kernel_launch(void* const* d_in, const int* in_sizes, int n_in,
                              void* d_out, int out_size, void* d_ws, size_t ws_size,
                              hipStream_t stream) {
  const float* x = (const float*)d_in[0];   // (8,1,512,512) f32
  const int*   y = (const int*)d_in[1];     // (8,1,512,512) i32 mask
  float* f        = (float*)d_ws;           // 2^21 f32 distance buffer (8 MB)
  float* partials = f + NTOT;               // 512 f32 block partials
  float* out      = (float*)d_out;          // scalar f32

  k_init_dt_b<<<HW / 256, 256, 0, stream>>>(y, f);
  k_dt_w<<<NB * NLINE, 256, 0, stream>>>(f);
  dim3 g3(NLINE / TW, NB);
  k_dt_h<<<g3, 256, 0, stream>>>(f);
  k_reduce_wmma<<<512, 256, 0, stream>>>(x, f, partials);
  k_final<<<1, 256, 0, stream>>>(partials, out);
}